// RepeatLengthRegulator_13511967113706
// MI455X (gfx1250) — compile-verified
//
#include <hip/hip_runtime.h>
#include <stdint.h>

// RepeatLengthRegulator: B=32, T=1024, D=768, MAX_DUR=8, PAD=0.0
#define B_C 32
#define T_C 1024
#define D_C 768
#define WAVES_PER_BLOCK 8
#define ROW_BYTES (D_C * 4)            // 3072 B per row
#define CHUNKS (ROW_BYTES / 512)       // 6 x (32 lanes * 16B) async ops per row

typedef float v4f __attribute__((ext_vector_type(4)));   // native vector: OK for
                                                         // __builtin_nontemporal_store

// Wave32 shuffle-up via LDS-unit lane permute (no LDS memory touched).
__device__ __forceinline__ int shfl_up_i32(int v, int off) {
  const int lane = (int)(threadIdx.x & 31);
  const int src  = (lane - off) & 31;
  return __builtin_amdgcn_ds_bpermute(src * 4, v);
}

// ---------------------------------------------------------------------------
// Kernel 1: per-batch-row inclusive scan of ds (wave-level Kogge-Stone via
// ds_bpermute + one 32-entry LDS pass for cross-wave offsets; 2 barriers),
// then scatter frame index t into idx_map[b, csum-d .. csum) and fill
// [total, L) with -1.  Exact inverse of searchsorted(csum, p, 'right').
// ---------------------------------------------------------------------------
__global__ __launch_bounds__(T_C) void scan_scatter_kernel(
    const int* __restrict__ ds, int* __restrict__ idx_map, int L) {
  __shared__ int wsum[T_C / 32];                 // 32 wave totals
  const int b    = blockIdx.x;
  const int t    = threadIdx.x;
  const int lane = t & 31;
  const int wv   = t >> 5;

  const int d = ds[b * T_C + t];

  // Intra-wave inclusive scan (5 shuffle steps on wave32).
  int x = d;
  #pragma unroll
  for (int off = 1; off < 32; off <<= 1) {
    int y = shfl_up_i32(x, off);
    if (lane >= off) x += y;
  }
  if (lane == 31) wsum[wv] = x;
  __syncthreads();

  // Wave 0 scans the 32 wave totals.
  if (wv == 0) {
    int w = wsum[lane];
    #pragma unroll
    for (int off = 1; off < 32; off <<= 1) {
      int y = shfl_up_i32(w, off);
      if (lane >= off) w += y;
    }
    wsum[lane] = w;
  }
  __syncthreads();

  const int csum  = x + (wv > 0 ? wsum[wv - 1] : 0);
  const int total = wsum[T_C / 32 - 1];
  int* __restrict__ row = idx_map + (size_t)b * (size_t)L;
  for (int k = csum - d; k < csum; ++k) row[k] = t;          // <= 8 writes
  for (int k = total + t; k < L; k += T_C) row[k] = -1;      // PAD region
}

// ---------------------------------------------------------------------------
// Kernel 2: one wave per output row. PAD rows store zero vectors (NT).
// Valid rows stream xs[b, idx, :] -> out[row, :] through LDS using the
// gfx1250 async global<->LDS path.  Stores are non-temporal so the 360MB
// write stream does not evict xs (96MB, fully L2-resident, re-read up to 8x).
// ---------------------------------------------------------------------------
__global__ __launch_bounds__(WAVES_PER_BLOCK * 32) void gather_copy_kernel(
    const float* __restrict__ xs, const int* __restrict__ idx_map,
    float* __restrict__ out, int L, int nrows) {
  __shared__ __align__(16) float buf[WAVES_PER_BLOCK * D_C];

  const int lane = threadIdx.x & 31;
  const int wid  = threadIdx.x >> 5;
  const int row  = blockIdx.x * WAVES_PER_BLOCK + wid;   // wave-uniform
  if (row >= nrows) return;

  const int idx = idx_map[row];
  float* __restrict__ dst = out + (size_t)row * D_C;

  if (idx < 0) {
    // PAD_VALUE = 0.0 — 16B per lane per chunk, non-temporal.
    const v4f z = {0.f, 0.f, 0.f, 0.f};
    #pragma unroll
    for (int c = 0; c < D_C / 128; ++c)
      __builtin_nontemporal_store(
          z, reinterpret_cast<v4f*>(dst + c * 128 + lane * 4));
    return;
  }

  const int b = row / L;
  const float* __restrict__ src = xs + ((size_t)b * T_C + (size_t)idx) * D_C;

  // Low 32 bits of a generic shared pointer == wave-relative LDS byte addr.
  const uint32_t la = (uint32_t)(uintptr_t)buf
                    + (uint32_t)(wid * ROW_BYTES) + (uint32_t)(lane * 16);
  const uint64_t sa = (uint64_t)(uintptr_t)src + (uint64_t)(lane * 16);

  // 6x async 128b loads (default RT hint: keep xs hot in L2); offset
  // advances both the global and the LDS address (ISA §10 pseudocode).
  #pragma unroll
  for (int c = 0; c < CHUNKS; ++c)
    asm volatile("global_load_async_to_lds_b128 %0, %1, off offset:%2"
                 :: "v"(la), "v"(sa), "i"(c * 512) : "memory");

  asm volatile("s_wait_asynccnt 0" ::: "memory");

  const uint64_t da = (uint64_t)(uintptr_t)dst + (uint64_t)(lane * 16);
  #pragma unroll
  for (int c = 0; c < CHUNKS; ++c)
    asm volatile("global_store_async_from_lds_b128 %0, %1, off offset:%2 th:TH_STORE_NT"
                 :: "v"(da), "v"(la), "i"(c * 512) : "memory");
  // S_ENDPGM performs an implicit wait-idle; no trailing wait needed.
}

// ---------------------------------------------------------------------------
extern "C" void kernel_launch(void* const* d_in, const int* in_sizes, int n_in,
                              void* d_out, int out_size, void* d_ws, size_t ws_size,
                              hipStream_t stream) {
  (void)in_sizes; (void)n_in; (void)ws_size;
  const float* xs = (const float*)d_in[0];
  const int*   ds = (const int*)d_in[1];
  float* out      = (float*)d_out;
  int*   idx_map  = (int*)d_ws;                    // B*L ints of scratch

  const int L = out_size / (B_C * D_C);            // harness fixes L via out size
  scan_scatter_kernel<<<B_C, T_C, 0, stream>>>(ds, idx_map, L);

  const int nrows  = B_C * L;
  const int blocks = (nrows + WAVES_PER_BLOCK - 1) / WAVES_PER_BLOCK;
  gather_copy_kernel<<<blocks, WAVES_PER_BLOCK * 32, 0, stream>>>(
      xs, idx_map, out, L, nrows);
}